// Mave_Global_Epistasis_Measurement_33809982554529
// MI455X (gfx1250) — compile-verified
//
#include <hip/hip_runtime.h>
#include <hip/hip_bf16.h>

typedef float v2f __attribute__((ext_vector_type(2)));
typedef float v8f __attribute__((ext_vector_type(8)));

__device__ __forceinline__ float fast_tanh(float x) {
#if __has_builtin(__builtin_amdgcn_tanhf)
    return __builtin_amdgcn_tanhf(x);
#else
    float r;
    asm volatile("v_tanh_f32 %0, %1" : "=v"(r) : "v"(x));
    return r;
#endif
}

__device__ __forceinline__ v2f shfl_xor_v2(v2f a, int mask) {
    v2f r;
    r.x = __shfl_xor(a.x, mask, 32);
    r.y = __shfl_xor(a.y, mask, 32);
    return r;
}

// Packed merge-reduce: p holds two row-partials {rowA, rowB}. Sums both over
// lane-bit `mask` (v_pk_add) and selects rowA's sum where (lane&mask)==0,
// rowB's where !=0.
__device__ __forceinline__ float merge_red2(v2f p, int mask, int lane) {
    v2f ps = p + shfl_xor_v2(p, mask);
    return (lane & mask) ? ps.y : ps.x;
}

// y[b] = a0 + sum_k bk[k] * tanh( sum_z z[b][z]*ck[k][z] + dk[k] )
// 16 batch rows per wave per tile via V_WMMA_F32_16X16X4_F32 (fp32 matrix path),
// tanh via v_tanh_f32, packed (v_pk_*) epilogue, merging butterfly reduction.
__global__ void __launch_bounds__(256)
epistasis_wmma_kernel(const float* __restrict__ z,
                      const float* __restrict__ a0p,
                      const float* __restrict__ bk,
                      const float* __restrict__ ck,
                      const float* __restrict__ dk,
                      float* __restrict__ out,
                      int nTiles)   // nTiles = B / 16
{
    const int lane    = threadIdx.x & 31;
    const int lane15  = lane & 15;         // column / row-in-half index
    const int hi      = lane >> 4;         // 0 = lanes 0-15, 1 = lanes 16-31
    // Force the loop induction into SGPRs: waveId is wave-uniform.
    const int waveId  = __builtin_amdgcn_readfirstlane(
                            (blockIdx.x * (blockDim.x >> 5)) + (threadIdx.x >> 5));
    const int nWaves  = gridDim.x * (blockDim.x >> 5);

    const float a0 = a0p[0];

    // ---- B-matrix fragments (ck^T), loaded once; ck is [K=64][Z=16] row-major.
    // WMMA B (4x16 f32, 2 VGPRs): VGPR j holds z-row (4i + j + 2*hi), col k = 16n+lane15.
    v2f bfrag[4][4];
    #pragma unroll
    for (int n = 0; n < 4; ++n) {
        const float* ckp = ck + (16 * n + lane15) * 16 + 2 * hi;
        #pragma unroll
        for (int i = 0; i < 4; ++i)
            bfrag[n][i] = *(const v2f*)(ckp + 4 * i);
    }

    // Per-lane epilogue constants, pre-splatted for packed math.
    v2f bk2[4], dk2[4];
    #pragma unroll
    for (int n = 0; n < 4; ++n) {
        const float b = bk[16 * n + lane15];
        const float d = dk[16 * n + lane15];
        bk2[n] = v2f{b, b};
        dk2[n] = v2f{d, d};
    }

    int tile = waveId;
    if (tile >= nTiles) return;

    // Scalar base pointers (SADDR addressing); per-lane offset is loop-invariant.
    const size_t laneOff  = (size_t)lane15 * 16 + 2 * hi;
    const size_t strideEl = (size_t)nWaves * 256;   // z elems per grid stride
    const float* zb = z + (size_t)tile * 256;
    float*       ob = out + ((size_t)tile << 4);

    // ---- First A-tile (16 rows x 16 z). A (16x4 f32): lanes 0-15 row=lane,
    // K pair {4i,4i+1}; lanes 16-31 K pair {4i+2,4i+3}. One NT float2 per step.
    v2f afrag[4];
    #pragma unroll
    for (int i = 0; i < 4; ++i)
        afrag[i] = __builtin_nontemporal_load((const v2f*)(zb + laneOff + 4 * i));

    while (true) {
        const bool hasNext = (tile + nWaves) < nTiles;

        // Branchless prefetch of the next tile (re-reads current tile on the
        // final iteration — always a valid address). Overlaps WMMA + epilogue.
        const float* zn = zb + (hasNext ? strideEl : 0);
        v2f anext[4];
        #pragma unroll
        for (int i = 0; i < 4; ++i)
            anext[i] = __builtin_nontemporal_load((const v2f*)(zn + laneOff + 4 * i));

        // ---- 16 chained WMMAs: acc[n] over z-steps i (interleaved for ILP).
        v8f acc[4] = {v8f{0.f,0.f,0.f,0.f,0.f,0.f,0.f,0.f},
                      v8f{0.f,0.f,0.f,0.f,0.f,0.f,0.f,0.f},
                      v8f{0.f,0.f,0.f,0.f,0.f,0.f,0.f,0.f},
                      v8f{0.f,0.f,0.f,0.f,0.f,0.f,0.f,0.f}};
        #pragma unroll
        for (int i = 0; i < 4; ++i) {
            #pragma unroll
            for (int n = 0; n < 4; ++n) {
                acc[n] = __builtin_amdgcn_wmma_f32_16x16x4_f32(
                    /*neg_a=*/false, afrag[i],
                    /*neg_b=*/false, bfrag[n][i],
                    /*c_mod=*/(short)0, acc[n],
                    /*reuse_a=*/false, /*reuse_b=*/false);
            }
        }

        // ---- Packed epilogue. C/D layout: VGPR r, lane l -> (M=r+8*hi, N=lane15).
        // p[rr] accumulates rows {2rr, 2rr+1} as a float2 (pk_add / pk_fma).
        v2f p[4];
        #pragma unroll
        for (int rr = 0; rr < 4; ++rr) {
            v2f s2 = v2f{0.f, 0.f};
            #pragma unroll
            for (int n = 0; n < 4; ++n) {
                v2f x = v2f{acc[n][2 * rr], acc[n][2 * rr + 1]} + dk2[n];
                v2f th = v2f{fast_tanh(x.x), fast_tanh(x.y)};
                s2 = s2 + bk2[n] * th;   // v_pk_fma_f32 / dual-FMA
            }
            p[rr] = s2;
        }

        // Packed merging butterfly: 8 row-partials -> 1 register.
        // After the tree, lane l holds the full sum for row (l&7) + 8*hi.
        float m0 = merge_red2(p[0], 1, lane);   // row  b0
        float m1 = merge_red2(p[1], 1, lane);   // row  2+b0
        float m2 = merge_red2(p[2], 1, lane);   // row  4+b0
        float m3 = merge_red2(p[3], 1, lane);   // row  6+b0
        float u0 = merge_red2(v2f{m0, m1}, 2, lane);   // row b0+2b1
        float u1 = merge_red2(v2f{m2, m3}, 2, lane);   // row 4+b0+2b1
        float f  = merge_red2(v2f{u0, u1}, 4, lane);   // row l&7
        f += __shfl_xor(f, 8, 32);

        const float y = a0 + f;
        if (lane15 < 8)
            __builtin_nontemporal_store(y, ob + (lane & 7) + 8 * hi);

        if (!hasNext) break;
        tile += nWaves;
        zb = zn;
        ob += (size_t)nWaves << 4;
        #pragma unroll
        for (int i = 0; i < 4; ++i) afrag[i] = anext[i];
    }
}

extern "C" void kernel_launch(void* const* d_in, const int* in_sizes, int n_in,
                              void* d_out, int out_size, void* d_ws, size_t ws_size,
                              hipStream_t stream) {
    const float* z  = (const float*)d_in[0];
    const float* a0 = (const float*)d_in[1];
    const float* bk = (const float*)d_in[2];
    const float* ck = (const float*)d_in[3];
    const float* dk = (const float*)d_in[4];
    float* out = (float*)d_out;

    const int Brows  = in_sizes[0] / 16;   // z is [B,16]
    const int nTiles = Brows / 16;

    dim3 block(256);
    dim3 grid(2048);                       // 16384 waves; grid-stride over tiles
    epistasis_wmma_kernel<<<grid, block, 0, stream>>>(z, a0, bk, ck, dk, out, nTiles);
}